// QSelfAttention_35880156791528
// MI455X (gfx1250) — compile-verified
//
#include <hip/hip_runtime.h>
#include <math.h>

typedef __attribute__((ext_vector_type(2))) float v2f;
typedef __attribute__((ext_vector_type(8))) float v8f;

#define S_LEN 512
#define HEADS 8
#define DHEAD 64
#define PITCH 68   // LDS row pitch in floats: conflict-free fragment reads, 16B-aligned rows

// ---------------------------------------------------------------------------
// CDNA5 async global->LDS copy (ASYNCcnt path), b128 per lane.
//   lds_off: raw LDS byte offset (low 32 bits of generic AS3 pointer)
//   gbase  : uniform 64-bit base (SGPR pair), voff: per-lane byte offset
// ---------------------------------------------------------------------------
__device__ __forceinline__ void async_ld_b128(unsigned lds_off,
                                              const float* gbase, unsigned voff)
{
    asm volatile("global_load_async_to_lds_b128 %0, %1, %2"
                 :: "v"(lds_off), "v"(voff),
                    "s"((unsigned long long)(size_t)gbase)
                 : "memory");
}
__device__ __forceinline__ void wait_async0()
{
    asm volatile("s_wait_asynccnt 0" ::: "memory");
}
__device__ __forceinline__ unsigned lds_off32(const void* p)
{
    return (unsigned)(size_t)p;   // generic AS3 address: low 32 bits == LDS offset
}

// ---------------------------------------------------------------------------
// Kernel 1: VQC projection (analytically reduced quantum circuit)
//   xin_q = W_in[q,:]·x + b_in[q];  theta_q = xin_q + w_q;  c_q = cos(theta_q)
//   z = [c1c2c3, c0c1, c0c1c2, c0c1c2c3];  out_d = W_out[d,:]·z + b_out[d]
// ---------------------------------------------------------------------------
__global__ __launch_bounds__(256) void proj_kernel(
    const float* __restrict__ Xq, const float* __restrict__ Xk, const float* __restrict__ Xv,
    const float* __restrict__ Wqin, const float* __restrict__ bqin,
    const float* __restrict__ Wkin, const float* __restrict__ bkin,
    const float* __restrict__ Wvin, const float* __restrict__ bvin,
    const float* __restrict__ wvq, const float* __restrict__ wvk, const float* __restrict__ wvv,
    const float* __restrict__ WqO, const float* __restrict__ bqO,
    const float* __restrict__ WkO, const float* __restrict__ bkO,
    const float* __restrict__ WvO, const float* __restrict__ bvO,
    float* __restrict__ Qp, float* __restrict__ Kp, float* __restrict__ Vp)
{
    const float *X, *Win, *binp, *wv, *Wout, *boutp; float* Out;
    if (blockIdx.y == 0)      { X=Xq; Win=Wqin; binp=bqin; wv=wvq; Wout=WqO; boutp=bqO; Out=Qp; }
    else if (blockIdx.y == 1) { X=Xk; Win=Wkin; binp=bkin; wv=wvk; Wout=WkO; boutp=bkO; Out=Kp; }
    else                      { X=Xv; Win=Wvin; binp=bvin; wv=wvv; Wout=WvO; boutp=bvO; Out=Vp; }

    __shared__ float sWin[256], sWout[256], sbout[64], sbw[8];
    const int t = threadIdx.x;
    sWin[t]  = Win[t];
    sWout[t] = Wout[t];
    if (t < 64) sbout[t] = boutp[t];
    if (t < 4)  { sbw[t] = binp[t]; sbw[4 + t] = wv[t]; }
    __syncthreads();

    const size_t m = (size_t)blockIdx.x * 256 + t;      // (b*S+s)*H+h
    const float4* x4 = (const float4*)(X + m * DHEAD);
    float a0 = sbw[0], a1 = sbw[1], a2 = sbw[2], a3 = sbw[3];
#pragma unroll
    for (int d4 = 0; d4 < 16; ++d4) {
        const float4 xv = x4[d4];
        const int d = 4 * d4;
        a0 = fmaf(xv.x, sWin[d],       fmaf(xv.y, sWin[d+1],       fmaf(xv.z, sWin[d+2],       fmaf(xv.w, sWin[d+3],       a0))));
        a1 = fmaf(xv.x, sWin[64+d],    fmaf(xv.y, sWin[64+d+1],    fmaf(xv.z, sWin[64+d+2],    fmaf(xv.w, sWin[64+d+3],    a1))));
        a2 = fmaf(xv.x, sWin[128+d],   fmaf(xv.y, sWin[128+d+1],   fmaf(xv.z, sWin[128+d+2],   fmaf(xv.w, sWin[128+d+3],   a2))));
        a3 = fmaf(xv.x, sWin[192+d],   fmaf(xv.y, sWin[192+d+1],   fmaf(xv.z, sWin[192+d+2],   fmaf(xv.w, sWin[192+d+3],   a3))));
    }
    const float c0 = cosf(a0 + sbw[4]);
    const float c1 = cosf(a1 + sbw[5]);
    const float c2 = cosf(a2 + sbw[6]);
    const float c3 = cosf(a3 + sbw[7]);
    const float z1 = c0 * c1;
    const float z2 = z1 * c2;
    const float z3 = z2 * c3;
    const float z0 = c1 * c2 * c3;

    float4* out4 = (float4*)(Out + m * DHEAD);
#pragma unroll
    for (int d4 = 0; d4 < 16; ++d4) {
        const int d = 4 * d4;
        float4 r;
        r.x = fmaf(z0, sWout[4*d+0],  fmaf(z1, sWout[4*d+1],  fmaf(z2, sWout[4*d+2],  fmaf(z3, sWout[4*d+3],  sbout[d]))));
        r.y = fmaf(z0, sWout[4*d+4],  fmaf(z1, sWout[4*d+5],  fmaf(z2, sWout[4*d+6],  fmaf(z3, sWout[4*d+7],  sbout[d+1]))));
        r.z = fmaf(z0, sWout[4*d+8],  fmaf(z1, sWout[4*d+9],  fmaf(z2, sWout[4*d+10], fmaf(z3, sWout[4*d+11], sbout[d+2]))));
        r.w = fmaf(z0, sWout[4*d+12], fmaf(z1, sWout[4*d+13], fmaf(z2, sWout[4*d+14], fmaf(z3, sWout[4*d+15], sbout[d+3]))));
        out4[d4] = r;
    }
}

// ---------------------------------------------------------------------------
// Kernel 2: flash attention per (b,h), 64 query rows per block, 4 waves.
//   softmax scale = 1/(D/2) = 1/32 (reproducing reference's D/2 scale).
//   V_WMMA_F32_16X16X4_F32 for S = Q·K^T and O += P·V; K/V tiles arrive via
//   global_load_async_to_lds_b128 (ASYNCcnt).
// ---------------------------------------------------------------------------
__global__ __launch_bounds__(128) void attn_kernel(
    const float* __restrict__ Qp, const float* __restrict__ Kp,
    const float* __restrict__ Vp, float* __restrict__ Ow)
{
    __shared__ alignas(16) float Klds[64 * PITCH];
    __shared__ alignas(16) float Vlds[64 * PITCH];
    __shared__ alignas(16) float Plds[4 * 16 * PITCH];

    const int t    = threadIdx.x;
    const int wave = t >> 5;
    const int lane = t & 31;
    const int hl   = lane >> 4;     // lane half: selects K-pair / row offset
    const int l16  = lane & 15;
    const int koff = hl * 2;

    const int idx = blockIdx.x;
    const int qb  = idx & 7;        // 8 q-blocks of 64
    const int bh  = idx >> 3;
    const int h   = bh & 7;
    const int b   = bh >> 3;
    const int q0  = qb * 64;

    // A-fragments of this wave's 16 Q rows (K = 64 -> 16 frags)
    v2f aq[16];
    {
        const int q = q0 + wave * 16 + l16;
        const float* qrow = Qp + ((size_t)((b * S_LEN + q) * HEADS + h)) * DHEAD;
#pragma unroll
        for (int tt = 0; tt < 16; ++tt) {
            aq[tt].x = qrow[4 * tt + koff];
            aq[tt].y = qrow[4 * tt + koff + 1];
        }
    }

    v8f o[4];
    float mrow[8], lrow[8];
#pragma unroll
    for (int j = 0; j < 4; ++j) { v8f z = {0.f,0.f,0.f,0.f,0.f,0.f,0.f,0.f}; o[j] = z; }
#pragma unroll
    for (int v = 0; v < 8; ++v) { mrow[v] = -3.0e38f; lrow[v] = 0.f; }

    for (int kt = 0; kt < 8; ++kt) {
        const int kv0 = kt * 64;
        __syncthreads();   // previous tile fully consumed before overwrite
        {
            const float* kbase = Kp + ((size_t)((b * S_LEN + kv0) * HEADS + h)) * DHEAD;
            const float* vbase = Vp + ((size_t)((b * S_LEN + kv0) * HEADS + h)) * DHEAD;
#pragma unroll
            for (int it = 0; it < 8; ++it) {
                const int i  = it * 128 + t;        // float4 index, 0..1023
                const int r  = i >> 4;
                const int c4 = i & 15;
                const unsigned voff = (unsigned)((r * (HEADS * DHEAD) + c4 * 4) * 4);
                async_ld_b128(lds_off32(&Klds[r * PITCH + c4 * 4]), kbase, voff);
                async_ld_b128(lds_off32(&Vlds[r * PITCH + c4 * 4]), vbase, voff);
            }
        }
        wait_async0();
        __syncthreads();

        // S_tile = Q(16x64) · K_tile^T(64x64)  -> 4 C-frags of 16x16
        v8f sfr[4];
#pragma unroll
        for (int j = 0; j < 4; ++j) {
            v8f c = {0.f,0.f,0.f,0.f,0.f,0.f,0.f,0.f};
            const int col = j * 16 + l16;
#pragma unroll
            for (int tt = 0; tt < 16; ++tt) {
                v2f bf;
                bf.x = Klds[col * PITCH + 4 * tt + koff];
                bf.y = Klds[col * PITCH + 4 * tt + koff + 1];
                c = __builtin_amdgcn_wmma_f32_16x16x4_f32(false, aq[tt], false, bf,
                                                          (short)0, c, false, false);
            }
            sfr[j] = c;
        }

        // online softmax update (scale 1/32)
#pragma unroll
        for (int v = 0; v < 8; ++v) {
            float mx = -3.0e38f;
#pragma unroll
            for (int j = 0; j < 4; ++j) {
                const float x = sfr[j][v] * 0.03125f;
                sfr[j][v] = x;
                mx = fmaxf(mx, x);
            }
#pragma unroll
            for (int mk = 1; mk < 16; mk <<= 1) mx = fmaxf(mx, __shfl_xor(mx, mk, 32));
            const float mnew = fmaxf(mrow[v], mx);

            float sum = 0.f;
#pragma unroll
            for (int j = 0; j < 4; ++j) {
                const float p = __expf(sfr[j][v] - mnew);
                sfr[j][v] = p;
                sum += p;
            }
#pragma unroll
            for (int mk = 1; mk < 16; mk <<= 1) sum += __shfl_xor(sum, mk, 32);
            const float corr = __expf(mrow[v] - mnew);
            lrow[v] = lrow[v] * corr + sum;
            mrow[v] = mnew;
#pragma unroll
            for (int j = 0; j < 4; ++j) o[j][v] *= corr;
        }

        // stage P (C layout -> row-major LDS), then read back as A-frags
        float* pw = Plds + wave * 16 * PITCH;
#pragma unroll
        for (int j = 0; j < 4; ++j)
#pragma unroll
            for (int v = 0; v < 8; ++v)
                pw[(v + 8 * hl) * PITCH + j * 16 + l16] = sfr[j][v];
        asm volatile("s_wait_dscnt 0" ::: "memory");

        // O += P(16x64) · V_tile(64x64)
#pragma unroll
        for (int tt = 0; tt < 16; ++tt) {
            v2f ap;
            ap.x = pw[l16 * PITCH + 4 * tt + koff];
            ap.y = pw[l16 * PITCH + 4 * tt + koff + 1];
#pragma unroll
            for (int j = 0; j < 4; ++j) {
                v2f bf;
                bf.x = Vlds[(4 * tt + koff)     * PITCH + j * 16 + l16];
                bf.y = Vlds[(4 * tt + koff + 1) * PITCH + j * 16 + l16];
                o[j] = __builtin_amdgcn_wmma_f32_16x16x4_f32(false, ap, false, bf,
                                                             (short)0, o[j], false, false);
            }
        }
    }

    // epilogue: O /= l, scatter to (B,S,H,D)
    const int qbase = q0 + wave * 16;
#pragma unroll
    for (int j = 0; j < 4; ++j)
#pragma unroll
        for (int v = 0; v < 8; ++v) {
            const int r = v + 8 * hl;
            const int d = j * 16 + l16;
            Ow[((size_t)((b * S_LEN + qbase + r) * HEADS + h)) * DHEAD + d] = o[j][v] / lrow[v];
        }
}

// ---------------------------------------------------------------------------
// Kernel 3: out = [Qf | O] (8192x1024) · Wo^T (1024x512) + bo, WMMA-f32 tiled,
//           tiles staged via global_load_async_to_lds_b128.
// ---------------------------------------------------------------------------
__global__ __launch_bounds__(128) void outproj_kernel(
    const float* __restrict__ Qp, const float* __restrict__ Ow,
    const float* __restrict__ Wo, const float* __restrict__ bo,
    float* __restrict__ out)
{
    __shared__ alignas(16) float Alds[64 * PITCH];
    __shared__ alignas(16) float Blds[64 * PITCH];

    const int t    = threadIdx.x;
    const int wave = t >> 5;
    const int lane = t & 31;
    const int hl   = lane >> 4;
    const int l16  = lane & 15;
    const int koff = hl * 2;

    const int m0 = (blockIdx.x & 127) * 64;   // 8192 rows / 64
    const int n0 = (blockIdx.x >> 7) * 64;    // 512 cols / 64

    v8f acc[4];
#pragma unroll
    for (int j = 0; j < 4; ++j) { v8f z = {0.f,0.f,0.f,0.f,0.f,0.f,0.f,0.f}; acc[j] = z; }

    for (int kt = 0; kt < 16; ++kt) {
        const int k0 = kt * 64;
        __syncthreads();
        {
            const float* abase = (k0 < 512) ? (Qp + (size_t)m0 * 512 + k0)
                                            : (Ow + (size_t)m0 * 512 + (k0 - 512));
            const float* bbase = Wo + (size_t)n0 * 1024 + k0;
#pragma unroll
            for (int it = 0; it < 8; ++it) {
                const int i  = it * 128 + t;
                const int r  = i >> 4;
                const int c4 = i & 15;
                async_ld_b128(lds_off32(&Alds[r * PITCH + c4 * 4]), abase,
                              (unsigned)((r * 512 + c4 * 4) * 4));
                async_ld_b128(lds_off32(&Blds[r * PITCH + c4 * 4]), bbase,
                              (unsigned)((r * 1024 + c4 * 4) * 4));
            }
        }
        wait_async0();
        __syncthreads();

#pragma unroll
        for (int tt = 0; tt < 16; ++tt) {
            v2f a;
            a.x = Alds[(wave * 16 + l16) * PITCH + 4 * tt + koff];
            a.y = Alds[(wave * 16 + l16) * PITCH + 4 * tt + koff + 1];
#pragma unroll
            for (int j = 0; j < 4; ++j) {
                v2f bf;
                bf.x = Blds[(j * 16 + l16) * PITCH + 4 * tt + koff];
                bf.y = Blds[(j * 16 + l16) * PITCH + 4 * tt + koff + 1];
                acc[j] = __builtin_amdgcn_wmma_f32_16x16x4_f32(false, a, false, bf,
                                                               (short)0, acc[j], false, false);
            }
        }
    }

#pragma unroll
    for (int j = 0; j < 4; ++j)
#pragma unroll
        for (int v = 0; v < 8; ++v) {
            const int row = m0 + wave * 16 + v + 8 * hl;
            const int col = n0 + j * 16 + l16;
            out[(size_t)row * 512 + col] = acc[j][v] + bo[col];
        }
}

// ---------------------------------------------------------------------------
extern "C" void kernel_launch(void* const* d_in, const int* in_sizes, int n_in,
                              void* d_out, int out_size, void* d_ws, size_t ws_size,
                              hipStream_t stream)
{
    (void)in_sizes; (void)n_in; (void)out_size; (void)ws_size;
    const float* q     = (const float*)d_in[0];
    const float* k     = (const float*)d_in[1];
    const float* v     = (const float*)d_in[2];
    // d_in[3]: mask — reference's masked_fill result is discarded; unused.
    const float* Wq_in = (const float*)d_in[4];
    const float* bq_in = (const float*)d_in[5];
    const float* Wk_in = (const float*)d_in[6];
    const float* bk_in = (const float*)d_in[7];
    const float* Wv_in = (const float*)d_in[8];
    const float* bv_in = (const float*)d_in[9];
    const float* wvq   = (const float*)d_in[10];
    const float* wvk   = (const float*)d_in[11];
    const float* wvv   = (const float*)d_in[12];
    const float* Wq    = (const float*)d_in[13];
    const float* bq    = (const float*)d_in[14];
    const float* Wk    = (const float*)d_in[15];
    const float* bk    = (const float*)d_in[16];
    const float* Wv    = (const float*)d_in[17];
    const float* bv    = (const float*)d_in[18];
    const float* Wo    = (const float*)d_in[19];
    const float* bo    = (const float*)d_in[20];
    float* out = (float*)d_out;

    float* Qp = (float*)d_ws;                 // 16 MB each
    float* Kp = Qp + 4194304;
    float* Vp = Kp + 4194304;
    float* Ob = Vp + 4194304;

    proj_kernel<<<dim3(256, 3), 256, 0, stream>>>(
        q, k, v, Wq_in, bq_in, Wk_in, bk_in, Wv_in, bv_in,
        wvq, wvk, wvv, Wq, bq, Wk, bk, Wv, bv, Qp, Kp, Vp);

    attn_kernel<<<dim3(1024), 128, 0, stream>>>(Qp, Kp, Vp, Ob);

    outproj_kernel<<<dim3(1024), 128, 0, stream>>>(Qp, Ob, Wo, bo, out);
}